// GAT_86569360818245
// MI455X (gfx1250) — compile-verified
//
#include <hip/hip_runtime.h>
#include <hip/hip_bf16.h>
#include <math.h>

#define N_NODES  16384
#define FEAT     128
#define EMB      64
#define N_EDGES  524288
#define MWORDS_PER_ROW (N_NODES / 32)             // 512 u32 words per adjacency row
#define MWORDS ((size_t)N_NODES * MWORDS_PER_ROW) // 8.39M words = 33.55 MB
#define L1_PER_ROW (MWORDS_PER_ROW / 32)          // 16 u32 hierarchy words per row
#define L1WORDS ((size_t)N_NODES * L1_PER_ROW)    // 262144 words = 1 MB
#define NEIGH_CAP 2048

typedef __attribute__((ext_vector_type(2))) float v2f;
typedef __attribute__((ext_vector_type(8))) float v8f;

__device__ __forceinline__ float leaky015(float x) { return x > 0.f ? x : 0.15f * x; }
__device__ __forceinline__ float elu1(float x)     { return x > 0.f ? x : expm1f(x); }

// ---------------------------------------------------------------- 1: zero scratch (mask + hierarchy + colsum)
__global__ void gat_zero(unsigned* __restrict__ mask, unsigned* __restrict__ lvl1,
                         float* __restrict__ colsum) {
    size_t i      = (size_t)blockIdx.x * blockDim.x + threadIdx.x;
    size_t stride = (size_t)gridDim.x * blockDim.x;
    for (size_t w = i; w < MWORDS; w += stride) mask[w] = 0u;
    for (size_t w = i; w < L1WORDS; w += stride) lvl1[w] = 0u;
    if (i < EMB) colsum[i] = 0.f;
}

// ---------------------------------------------------------------- 2: COO edges -> 2-level adjacency bitmask
// Level 0: one bit per (i,j) pair (dedup for free: matches adj > 0 exactly).
// Level 1: one bit per nonzero level-0 word, so the row kernel skips empty words.
__global__ void gat_build_mask(const int* __restrict__ A, unsigned* __restrict__ mask,
                               unsigned* __restrict__ lvl1) {
    int e = blockIdx.x * blockDim.x + threadIdx.x;
    if (e >= N_EDGES) return;
    int r = A[e];
    int c = A[N_EDGES + e];
    int w = c >> 5;                                   // which mask word within the row
    atomicOr(&mask[(size_t)r * MWORDS_PER_ROW + w], 1u << (c & 31));
    atomicOr(&lvl1[(size_t)r * L1_PER_ROW + (w >> 5)], 1u << (w & 31));
}

// ---------------------------------------------------------------- 3: Wh = X @ Ws via V_WMMA_F32_16X16X4_F32
// One wave32 per 16x16 output tile. 4 waves/block cover the 4 column tiles (EMB=64).
__global__ void __launch_bounds__(128) gat_gemm_wh(const float* __restrict__ X,
                                                   const float* __restrict__ Ws,
                                                   float* __restrict__ Wh) {
    const int lane = threadIdx.x & 31;
    const int wave = threadIdx.x >> 5;      // column tile 0..3
    const int rt   = blockIdx.x;            // row tile 0..1023
    const int mn   = lane & 15;             // M index for A, N index for B/C/D
    const int half = lane >> 4;             // lane group selects K pair / M+8
    const int col  = wave * 16 + mn;

    const float* xrow = X + ((size_t)(rt * 16 + mn)) * FEAT;
    v8f c = {};
#pragma unroll 4
    for (int k0 = 0; k0 < FEAT; k0 += 4) {
        const int kb = k0 + half * 2;       // lanes 0-15: K={k0,k0+1}; lanes 16-31: K={k0+2,k0+3}
        v2f a, b;
        a.x = xrow[kb];
        a.y = xrow[kb + 1];
        b.x = Ws[(size_t)kb * EMB + col];
        b.y = Ws[(size_t)(kb + 1) * EMB + col];
        c = __builtin_amdgcn_wmma_f32_16x16x4_f32(false, a, false, b, (short)0, c,
                                                  false, false);
    }
#pragma unroll
    for (int v = 0; v < 8; ++v) {
        const int row = rt * 16 + v + half * 8;   // C/D layout: VGPR v -> M = v + 8*half
        Wh[(size_t)row * EMB + col] = c[v];
    }
}

// ---------------------------------------------------------------- 4: s1 = Wh . a[:64], s2 = Wh . a[64:]
__global__ void gat_scores(const float* __restrict__ Wh, const float* __restrict__ a,
                           float* __restrict__ s1, float* __restrict__ s2) {
    int i = blockIdx.x * blockDim.x + threadIdx.x;
    if (i >= N_NODES) return;
    const float* row = Wh + (size_t)i * EMB;
    float acc1 = 0.f, acc2 = 0.f;
#pragma unroll 8
    for (int c = 0; c < EMB; ++c) {
        float w = row[c];
        acc1 += w * a[c];
        acc2 += w * a[EMB + c];
    }
    s1[i] = acc1;
    s2[i] = acc2;
}

// ---------------------------------------------------------------- 5: column sums of Wh (empty-row uniform-softmax fallback)
__global__ void gat_colsum(const float* __restrict__ Wh, float* __restrict__ colsum) {
    const int c = threadIdx.x;              // 64 threads = 64 columns
    const int b = blockIdx.x;               // 256 blocks x 64 rows each
    float p = 0.f;
    for (int r = b * 64; r < (b + 1) * 64; ++r) p += Wh[(size_t)r * EMB + c];
    atomicAdd(&colsum[c], p);
}

// ---------------------------------------------------------------- 6: per-row sparse masked softmax + aggregation + ELU
__global__ void __launch_bounds__(64) gat_row(const unsigned* __restrict__ mask,
                                              const unsigned* __restrict__ lvl1,
                                              const float* __restrict__ Wh,
                                              const float* __restrict__ s1,
                                              const float* __restrict__ s2,
                                              const float* __restrict__ colsum,
                                              float* __restrict__ out) {
    __shared__ int      list[NEIGH_CAP];
    __shared__ int      cnt;
    __shared__ float    red[64];
    __shared__ unsigned h[L1_PER_ROW];
    const int i = blockIdx.x;
    const int t = threadIdx.x;
    if (t == 0) cnt = 0;
    if (t < L1_PER_ROW) h[t] = lvl1[(size_t)i * L1_PER_ROW + t];
    __syncthreads();

    // Gather set bits of this row's adjacency bitmask into an LDS neighbor list,
    // touching only level-0 words the hierarchy marks nonzero (~32 of 512).
    const unsigned* mrow = mask + (size_t)i * MWORDS_PER_ROW;
    for (int w = t; w < MWORDS_PER_ROW; w += 64) {
        if ((h[w >> 5] >> (w & 31)) & 1u) {
            unsigned bits = mrow[w];
            while (bits) {
                int b = __ffs(bits) - 1;
                bits &= bits - 1u;
                int idx = atomicAdd(&cnt, 1);
                if (idx < NEIGH_CAP) list[idx] = w * 32 + b;
            }
        }
    }
    __syncthreads();
    int nn = cnt;
    if (nn > NEIGH_CAP) nn = NEIGH_CAP;

    if (nn == 0) {
        // Dense reference: all-NEG_INF row -> softmax is uniform 1/N -> mean of Wh columns.
        out[(size_t)i * EMB + t] = elu1(colsum[t] * (1.0f / N_NODES));
        return;
    }

    const float si = s1[i];

    // Row max of leaky_relu(s1_i + s2_j) over the neighbor set.
    float mx = -3.4e38f;
    for (int k = t; k < nn; k += 64) mx = fmaxf(mx, leaky015(si + s2[list[k]]));
    red[t] = mx;
    __syncthreads();
    for (int s = 32; s > 0; s >>= 1) {
        if (t < s) red[t] = fmaxf(red[t], red[t + s]);
        __syncthreads();
    }
    const float m = red[0];

    // Each thread owns one output channel; coalesced Wh[j][0..63] reads across the block.
    // exp(NEG_INF - m) == 0 in fp32, so the sparse sum equals the dense masked softmax.
    float acc = 0.f, Z = 0.f;
    for (int k = 0; k < nn; ++k) {
        const int j = list[k];
        const float w = __expf(leaky015(si + s2[j]) - m);
        Z   += w;
        acc += w * Wh[(size_t)j * EMB + t];
    }
    out[(size_t)i * EMB + t] = elu1(acc / Z);
}

// ----------------------------------------------------------------
extern "C" void kernel_launch(void* const* d_in, const int* in_sizes, int n_in,
                              void* d_out, int out_size, void* d_ws, size_t ws_size,
                              hipStream_t stream) {
    const int*   A  = (const int*)d_in[0];     // [2, N_EDGES] COO
    const float* X  = (const float*)d_in[1];   // [N, 128]
    const float* Ws = (const float*)d_in[2];   // [128, 64]
    const float* a  = (const float*)d_in[3];   // [128]
    float* out = (float*)d_out;                // [N, 64]

    // Scratch layout inside d_ws (~38.8 MB total)
    unsigned* mask   = (unsigned*)d_ws;                         // 33.55 MB level-0 bitmask
    unsigned* lvl1   = mask + MWORDS;                           // 1 MB level-1 hierarchy
    float*    Wh     = (float*)(lvl1 + L1WORDS);                // 4 MB
    float*    s1     = Wh + (size_t)N_NODES * EMB;              // 64 KB
    float*    s2     = s1 + N_NODES;                            // 64 KB
    float*    colsum = s2 + N_NODES;                            // 256 B

    gat_zero      <<<4096, 256, 0, stream>>>(mask, lvl1, colsum);
    gat_build_mask<<<N_EDGES / 256, 256, 0, stream>>>(A, mask, lvl1);
    gat_gemm_wh   <<<N_NODES / 16, 128, 0, stream>>>(X, Ws, Wh);
    gat_scores    <<<N_NODES / 256, 256, 0, stream>>>(Wh, a, s1, s2);
    gat_colsum    <<<256, 64, 0, stream>>>(Wh, colsum);
    gat_row       <<<N_NODES, 64, 0, stream>>>(mask, lvl1, Wh, s1, s2, colsum, out);
}